// NNSK_59923383713794
// MI455X (gfx1250) — compile-verified
//
#include <hip/hip_runtime.h>
#include <hip/hip_bf16.h>
#include <stdint.h>

// NNSK edge/node feature expansion for MI455X (gfx1250).
// Store-bandwidth-bound elementwise workload: no GEMM structure -> no WMMA.
// CDNA5 paths: async global->LDS b128 fill (ASYNCcnt + s_wait_asynccnt) for
// the parameter table, LDS-staged fully-coalesced b128 non-temporal stores.

#define R_ELEM 14
#define TPB 256

typedef float v4f __attribute__((ext_vector_type(4)));

// ---- workspace table layout (floats) ----
// [0..55]    H_A1  : symmetrized hopping alpha1          (4 x 14)
// [56..111]  H_A2  : |symmetrized hopping alpha2|        (4 x 14)
// [112..167] O_A1  : symmetrized overlap alpha1 + same*SYM_MASK
// [168..223] O_A2  : |symmetrized overlap alpha2|
// [224..227] R0    : 0.5*(bll[p0-1]+bll[p1-1]) per bond type
#define TAB_WORDS 228
#define TAB_VEC4  57   // 228/4 b128 async transfers

__global__ void nnsk_prep(const float* __restrict__ hp,
                          const float* __restrict__ op,
                          const float* __restrict__ bll,
                          float* __restrict__ tab) {
    int t = threadIdx.x;
    if (t < 56) {
        int b = t / R_ELEM;
        int r = t - b * R_ELEM;
        // SYM_MASK[{0,4,5,11,12,13}] = 1
        float m = (r == 0 || r == 4 || r == 5 || r == 11 || r == 12 || r == 13)
                      ? 1.0f : 0.0f;
        // REFLECTIVE_BONDS = [0,2,1,3]
        int rb = (b == 1) ? 2 : ((b == 2) ? 1 : b);
        int i0 = b * (R_ELEM * 2) + r * 2;
        int i1 = rb * (R_ELEM * 2) + r * 2;

        float h0 = hp[i0],     h0r = hp[i1];
        float h1 = hp[i0 + 1], h1r = hp[i1 + 1];
        float sh0 = (m > 0.0f) ? 0.5f * (h0 + h0r) : h0;
        float sh1 = (m > 0.0f) ? 0.5f * (h1 + h1r) : h1;

        float o0 = op[i0],     o0r = op[i1];
        float o1 = op[i0 + 1], o1r = op[i1 + 1];
        float so0 = (m > 0.0f) ? 0.5f * (o0 + o0r) : o0;
        float so1 = (m > 0.0f) ? 0.5f * (o1 + o1r) : o1;

        // same_elem: BOND_TO_PAIR rows 0:(14,14) 1:(14,8) 2:(8,14) 3:(8,8)
        float same = (b == 0 || b == 3) ? 1.0f : 0.0f;

        tab[t]        = sh0;
        tab[56 + t]   = fabsf(sh1);
        tab[112 + t]  = so0 + same * m;   // fold paraconst into alpha1
        tab[168 + t]  = fabsf(so1);
    } else if (t < 60) {
        int b = t - 56;
        const int p0[4] = {14, 14, 8, 8};
        const int p1[4] = {14, 8, 14, 8};
        tab[224 + b] = 0.5f * (bll[p0[b] - 1] + bll[p1[b] - 1]);
    }
}

// Coalesced block store of TPB*R_ELEM staged floats; b128 fast path when the
// whole block is in range and the destination segment is 16B aligned.
__device__ __forceinline__ void store_block(const float* __restrict__ s_stage,
                                            float* __restrict__ dstf,
                                            size_t base, size_t lim,
                                            int tid, bool full) {
    if (full && ((((size_t)(dstf + base)) & 15u) == 0)) {
        v4f* __restrict__ dst = (v4f*)(dstf + base);
        const v4f* __restrict__ src = (const v4f*)s_stage;
#pragma unroll
        for (int k = 0; k < 3; ++k)
            __builtin_nontemporal_store(src[tid + k * TPB], dst + tid + k * TPB);
        if (tid < (TPB * R_ELEM / 4 - 3 * TPB))   // 896 - 768 = 128 lanes
            __builtin_nontemporal_store(src[tid + 3 * TPB], dst + tid + 3 * TPB);
    } else {
#pragma unroll
        for (int k = 0; k < R_ELEM; ++k) {
            size_t idx = base + tid + (size_t)k * TPB;
            if (idx < lim)
                __builtin_nontemporal_store(s_stage[tid + k * TPB], dstf + idx);
        }
    }
}

__global__ __launch_bounds__(TPB) void nnsk_edge(
    const float* __restrict__ rij_arr, const int* __restrict__ et_arr,
    const float* __restrict__ tab, float* __restrict__ out_h,
    float* __restrict__ out_o, int n) {
    __shared__ __align__(16) float s_tab[TAB_WORDS + 4];
    __shared__ __align__(16) float s_stage[TPB * R_ELEM];

    const int tid = threadIdx.x;

    // --- CDNA5 async global->LDS b128 fill of the 912-byte parameter table ---
    if (tid < TAB_VEC4) {
        unsigned           laddr = (unsigned)(size_t)&s_tab[tid * 4];  // LDS byte addr = low 32b of generic ptr
        unsigned long long gaddr = (unsigned long long)(size_t)(tab + tid * 4);
        asm volatile("global_load_async_to_lds_b128 %0, %1, off"
                     :: "v"(laddr), "v"(gaddr) : "memory");
    }
    asm volatile("s_wait_asynccnt 0" ::: "memory");
    __syncthreads();

    const int e = blockIdx.x * TPB + tid;
    float L2 = 0.0f, c = 0.0f;
    int et = 0;
    if (e < n) {
        float rij = __builtin_nontemporal_load(rij_arr + e);
        et        = __builtin_nontemporal_load(et_arr + e);
        float r0  = s_tab[224 + et];
        // (r0/rij)^(1+a2) = exp2(log2(r0/rij) * (1+a2)); v_log_f32 is log2
        L2 = __builtin_amdgcn_logf(r0 * __builtin_amdgcn_rcpf(rij));
        // fcut = 1/(1+exp((rij-6)/0.2)) = 1/(1+exp2((rij-6)*5*log2(e)))
        float ex = __builtin_amdgcn_exp2f((rij - 6.0f) * 7.21347520444481703680f);
        c = __builtin_amdgcn_rcpf(1.0f + ex);
    }

    const size_t base = (size_t)blockIdx.x * (TPB * R_ELEM);
    const size_t lim  = (size_t)n * R_ELEM;
    const bool   full = (blockIdx.x + 1) * TPB <= n;

    // ---- pass 1: hopping ----
    {
        const float* A1 = s_tab + et * R_ELEM;
        const float* A2 = s_tab + 56 + et * R_ELEM;
#pragma unroll
        for (int r = 0; r < R_ELEM; ++r) {
            float v = (A1[r] * c) * __builtin_amdgcn_exp2f(L2 * (1.0f + A2[r]));
            s_stage[tid * R_ELEM + r] = v;
        }
    }
    __syncthreads();
    store_block(s_stage, out_h, base, lim, tid, full);
    __syncthreads();

    // ---- pass 2: overlap (paraconst already folded into O_A1) ----
    {
        const float* B1 = s_tab + 112 + et * R_ELEM;
        const float* B2 = s_tab + 168 + et * R_ELEM;
#pragma unroll
        for (int r = 0; r < R_ELEM; ++r) {
            float v = (B1[r] * c) * __builtin_amdgcn_exp2f(L2 * (1.0f + B2[r]));
            s_stage[tid * R_ELEM + r] = v;
        }
    }
    __syncthreads();
    store_block(s_stage, out_o, base, lim, tid, full);
}

__global__ void nnsk_node(const int* __restrict__ atype,
                          const float* __restrict__ onsite_param,
                          const float* __restrict__ onsite_E_base,
                          float* __restrict__ out, int n3) {
    int idx = blockIdx.x * TPB + threadIdx.x;
    if (idx >= n3) return;
    int i = idx / 3;
    int k = idx - i * 3;
    int t = atype[i];
    int z = (t == 0) ? 14 : 8;  // Z_OF_TYPE = [14, 8]
    float v = onsite_E_base[(z - 1) * 3 + k] + onsite_param[t * 3 + k];
    __builtin_nontemporal_store(v, out + idx);
}

extern "C" void kernel_launch(void* const* d_in, const int* in_sizes, int n_in,
                              void* d_out, int out_size, void* d_ws, size_t ws_size,
                              hipStream_t stream) {
    const float* hp            = (const float*)d_in[0];
    const float* op            = (const float*)d_in[1];
    const float* onsite_param  = (const float*)d_in[2];
    const float* onsite_E_base = (const float*)d_in[3];
    const float* bll           = (const float*)d_in[4];
    const float* edge_length   = (const float*)d_in[5];
    const int*   edge_type     = (const int*)d_in[6];
    const int*   atom_type     = (const int*)d_in[7];

    const int n_edges = in_sizes[5];
    const int n_nodes = in_sizes[7];

    float* out   = (float*)d_out;
    float* out_h = out;
    float* out_o = out + (size_t)n_edges * R_ELEM;
    float* out_n = out + (size_t)n_edges * (2 * R_ELEM);
    float* tab   = (float*)d_ws;

    nnsk_prep<<<1, 64, 0, stream>>>(hp, op, bll, tab);

    int nb = (n_edges + TPB - 1) / TPB;
    nnsk_edge<<<nb, TPB, 0, stream>>>(edge_length, edge_type, tab,
                                      out_h, out_o, n_edges);

    int n3 = n_nodes * 3;
    nnsk_node<<<(n3 + TPB - 1) / TPB, TPB, 0, stream>>>(
        atom_type, onsite_param, onsite_E_base, out_n, n3);
}